// NMS_50508815401507
// MI455X (gfx1250) — compile-verified
//
#include <hip/hip_runtime.h>
#include <hip/hip_bf16.h>

// NMS on MI455X (gfx1250): one workgroup per batch row.
//   - 256KB score plane lives in LDS (CDNA5 WGP has 320KB).
//   - wave32 shuffle reductions for argmax (warpSize==32 on gfx1250).
//   - phase-1 staging via GLOBAL_LOAD_ASYNC_TO_LDS_B32 (builtin confirmed to
//     exist on this toolchain; it takes typed AS1/AS3 int* pointers).
//   - suppression pass fused with next iteration's local argmax (disjoint
//     per-thread index ownership -> no extra barrier).

#define NMS_BLOCK 1024
#define NMS_SCORE_TH 0.3f
#define NMS_DIST_TH2 256.0f   // 16.0^2

#if defined(__has_builtin)
#  if __has_builtin(__builtin_amdgcn_global_load_async_to_lds_b32) && \
      __has_builtin(__builtin_amdgcn_s_wait_asynccnt)
#    define NMS_ASYNC_LDS 1
#  endif
#  if __has_builtin(__builtin_amdgcn_s_wait_tensorcnt)
#    define NMS_HAVE_TENSORCNT 1
#  endif
#endif

typedef __attribute__((address_space(1))) int* nms_gptr_i32;
typedef __attribute__((address_space(3))) int* nms_lptr_i32;

extern "C" __global__ __launch_bounds__(NMS_BLOCK)
void nms_gfx1250_kernel(const float* __restrict__ preds,
                        const int* __restrict__ maxb_p,
                        float* __restrict__ out,
                        int n) {
  extern __shared__ float smem[];
  float* s_score = smem;                      // n floats (256KB for n=65536)
  float* s_wval  = smem + n;                  // 32 per-wave maxima
  int*   s_widx  = (int*)(smem + n + 32);     // 32 per-wave argmax indices
  float* s_best  = smem + n + 64;             // [0]=score [1]=idx(bits) [2]=qx [3]=qy

  const int tid  = threadIdx.x;
  const int bb   = blockIdx.x;
  const int maxb = maxb_p[0];
  const float* base = preds + (size_t)bb * (size_t)n * 8u;
  float* outb = out + (size_t)bb * (size_t)maxb * 8u;

  // ---- Phase 1: stage thresholded scores into LDS, seed local argmax ----
  float lm = -1.0f;
  int   li = 0;
#ifdef NMS_ASYNC_LDS
  for (int i = tid; i < n; i += NMS_BLOCK) {
    // async copy of the strided score column (element 2 of each 32B row)
    __builtin_amdgcn_global_load_async_to_lds_b32(
        (nms_gptr_i32)(base + (size_t)i * 8u + 2u),
        (nms_lptr_i32)(s_score + i),
        /*imm offset*/ 0, /*cpol*/ 0);
  }
  __builtin_amdgcn_s_wait_asynccnt(0);
  for (int i = tid; i < n; i += NMS_BLOCK) {   // threshold in place
    float v = s_score[i];
    v = (v >= NMS_SCORE_TH) ? v : 0.0f;
    s_score[i] = v;
    if (v > lm) { lm = v; li = i; }
  }
#else
  for (int i = tid; i < n; i += NMS_BLOCK) {
    float v = base[(size_t)i * 8u + 2u];
    v = (v >= NMS_SCORE_TH) ? v : 0.0f;
    s_score[i] = v;
    if (v > lm) { lm = v; li = i; }
  }
#endif

  const int wid  = tid >> 5;
  const int lane = tid & 31;

  // ---- Phase 2: greedy select + suppress, maxb times ----
  for (int it = 0; it < maxb; ++it) {
    // wave32 reduction of (lm, li); ties -> lower index (matches stable argsort)
#pragma unroll
    for (int off = 16; off > 0; off >>= 1) {
      float ov = __shfl_xor(lm, off, 32);
      int   oi = __shfl_xor(li, off, 32);
      if (ov > lm || (ov == lm && oi < li)) { lm = ov; li = oi; }
    }
    if (lane == 0) { s_wval[wid] = lm; s_widx[wid] = li; }
    __syncthreads();

    if (wid == 0) {
      float v  = s_wval[lane];
      int   i2 = s_widx[lane];
#pragma unroll
      for (int off = 16; off > 0; off >>= 1) {
        float ov = __shfl_xor(v, off, 32);
        int   oi = __shfl_xor(i2, off, 32);
        if (ov > v || (ov == v && oi < i2)) { v = ov; i2 = oi; }
      }
      if (lane == 0) {
        s_best[0] = v;
        ((int*)s_best)[1] = i2;
        if (v > 0.0f) {
          const float4* row = (const float4*)(base + (size_t)i2 * 8u);
          float4 r0 = row[0];
          float4 r1 = row[1];
          float4* orow = (float4*)(outb + (size_t)it * 8u);
          orow[0] = r0;
          orow[1] = r1;
          s_best[2] = r0.x;   // qx
          s_best[3] = r0.y;   // qy
        }
      }
    }
    __syncthreads();

    float bs = s_best[0];
    if (bs <= 0.0f) {
      // nothing left: emit zero rows for [it, maxb) (reference yields zeros)
      for (int j = it * 8 + tid; j < maxb * 8; j += NMS_BLOCK) outb[j] = 0.0f;
      break;
    }
    const float qx = s_best[2];
    const float qy = s_best[3];

    // fused suppression + next local argmax over this thread's own indices
    lm = -1.0f;
    li = 0;
    for (int i = tid; i < n; i += NMS_BLOCK) {
      float v = s_score[i];
      if (v != 0.0f) {
        float2 xy = *(const float2*)(base + (size_t)i * 8u);
        float dx = xy.x - qx;
        float dy = xy.y - qy;
        if (dx * dx + dy * dy < NMS_DIST_TH2) { v = 0.0f; s_score[i] = 0.0f; }
      }
      if (v > lm) { lm = v; li = i; }
    }
    // no barrier needed: each thread reads/writes only its own score slots;
    // reduction scratch is fenced by the two barriers at the top of the loop.
  }

#ifdef NMS_HAVE_TENSORCNT
  __builtin_amdgcn_s_wait_tensorcnt(0);  // probe-confirmed gfx1250 op; no-op here
#endif
}

extern "C" void kernel_launch(void* const* d_in, const int* in_sizes, int n_in,
                              void* d_out, int out_size, void* d_ws, size_t ws_size,
                              hipStream_t stream) {
  (void)n_in; (void)out_size; (void)d_ws; (void)ws_size;
  const float* preds = (const float*)d_in[0];
  const int*   maxb  = (const int*)d_in[1];
  float*       out   = (float*)d_out;

  const int N = 65536, V = 8;
  int B = in_sizes[0] / (N * V);
  if (B < 1) B = 1;

  // dynamic LDS: N scores + reduction scratch (+ slack)
  size_t shmem = (size_t)N * sizeof(float) + 512;

  nms_gfx1250_kernel<<<dim3(B), dim3(NMS_BLOCK), shmem, stream>>>(preds, maxb, out, N);
}